// AttentionLayer_48722109006189
// MI455X (gfx1250) — compile-verified
//
#include <hip/hip_runtime.h>
#include <math.h>

typedef __attribute__((ext_vector_type(2))) float v2f;
typedef __attribute__((ext_vector_type(8))) float v8f;

#define B_   8
#define N_   256
#define FIN  64
#define HID_ 128
#define EDGEF 64
#define NEG_SLOPE_ 0.2f

// ---------------------------------------------------------------------------
// Kernel 1: per-node scalars  s_i = a_src . leaky(W x_i),  d_i = a_dst . leaky(W x_i),
//           q_i = ||x_i||^2.  One thread per node; W_lin staged in LDS.
// ---------------------------------------------------------------------------
__global__ __launch_bounds__(256) void gat_node_stats(
    const float* __restrict__ src, const float* __restrict__ W_lin,
    const float* __restrict__ a_src, const float* __restrict__ a_dst,
    float* __restrict__ sArr, float* __restrict__ dArr, float* __restrict__ qArr)
{
    __shared__ float Wl[HID_ * FIN];              // 32 KB
    for (int t = threadIdx.x; t < HID_ * FIN; t += blockDim.x) Wl[t] = W_lin[t];
    __syncthreads();

    const int idx = blockIdx.x * blockDim.x + threadIdx.x;   // 0 .. B*N-1
    const float* xr = src + (size_t)idx * FIN;

    float r[FIN];
    float q = 0.f;
#pragma unroll
    for (int f = 0; f < FIN; ++f) { r[f] = xr[f]; q = fmaf(r[f], r[f], q); }

    float s = 0.f, d = 0.f;
    for (int h = 0; h < HID_; ++h) {
        float acc = 0.f;
#pragma unroll
        for (int f = 0; f < FIN; ++f) acc = fmaf(Wl[h * FIN + f], r[f], acc);
        acc = acc > 0.f ? acc : acc * NEG_SLOPE_;             // leaky_relu(0.2)
        s = fmaf(acc, a_src[h], s);
        d = fmaf(acc, a_dst[h], d);
    }
    sArr[idx] = s; dArr[idx] = d; qArr[idx] = q;
}

// ---------------------------------------------------------------------------
// Kernel 2: per (batch, 16-row strip):
//   G = S S^T via v_wmma_f32_16x16x4_f32 (K=64 -> 16 chained WMMAs per tile),
//   e[i,j] = (s_i + d_j + coef*sqrt(max(q_i+q_j-2G,0))*adj) * mask,
//   row softmax in-wave (wave32 shuffles).
// Block = 256 threads = 8 wave32; wave w owns column tiles 2w, 2w+1.
// ---------------------------------------------------------------------------
__global__ __launch_bounds__(256) void gat_attn_softmax(
    const float* __restrict__ src, const int* __restrict__ adj, const int* __restrict__ mask,
    const float* __restrict__ W_edge, const float* __restrict__ a_edge,
    const float* __restrict__ sArr, const float* __restrict__ dArr, const float* __restrict__ qArr,
    float* __restrict__ out)
{
    __shared__ float ebuf[16][N_];                // 16 KB strip of e-values
    __shared__ float sRow[16], qRow[16];
    __shared__ float dCol[N_], qCol[N_];

    const int b     = blockIdx.y;
    const int iBase = blockIdx.x * 16;
    const int tid   = threadIdx.x;

    if (tid < 16) {
        sRow[tid] = sArr[b * N_ + iBase + tid];
        qRow[tid] = qArr[b * N_ + iBase + tid];
    }
    dCol[tid] = dArr[b * N_ + tid];
    qCol[tid] = qArr[b * N_ + tid];

    // coef = dot(W_edge, a_edge)  (wave-uniform scalar loads, negligible)
    float coef = 0.f;
#pragma unroll
    for (int k = 0; k < EDGEF; ++k) coef = fmaf(W_edge[k], a_edge[k], coef);
    __syncthreads();

    const int wave   = tid >> 5;
    const int lane   = tid & 31;
    const int lane15 = lane & 15;
    const int hiHalf = lane >> 4;                  // 0: lanes 0-15, 1: lanes 16-31
    const int kOff   = hiHalf << 1;                // +2 K-offset for high half

    // A fragments: 16 rows of this strip, all K=64 held in registers (16 x v2f)
    const float* Sb      = src + (size_t)b * N_ * FIN;
    const float* aRowPtr = Sb + (size_t)(iBase + lane15) * FIN + kOff;
    v2f afrag[16];
#pragma unroll
    for (int kk = 0; kk < 16; ++kk) {
        afrag[kk].x = aRowPtr[4 * kk + 0];
        afrag[kk].y = aRowPtr[4 * kk + 1];
    }

    for (int t = 0; t < 2; ++t) {
        const int colBase   = (wave * 2 + t) * 16;
        const float* bColPtr = Sb + (size_t)(colBase + lane15) * FIN + kOff;

        v8f c = {0.f, 0.f, 0.f, 0.f, 0.f, 0.f, 0.f, 0.f};
#pragma unroll
        for (int kk = 0; kk < 16; ++kk) {
            v2f bfrag;
            bfrag.x = bColPtr[4 * kk + 0];
            bfrag.y = bColPtr[4 * kk + 1];
            c = __builtin_amdgcn_wmma_f32_16x16x4_f32(
                    false, afrag[kk], false, bfrag, (short)0, c, false, false);
        }

        // Epilogue directly on the C-fragment layout:
        // VGPR r, lanes 0-15 -> M=r ; lanes 16-31 -> M=r+8 ; N = lane&15
        const int j = colBase + lane15;
#pragma unroll
        for (int r = 0; r < 8; ++r) {
            const int iLoc = r + hiHalf * 8;
            const int i    = iBase + iLoc;
            const float g  = c[r];
            float sq  = qRow[iLoc] + qCol[j] - 2.f * g;
            float nrm = sq > 0.f ? sqrtf(sq) : 0.f;
            float av  = (float)adj [((size_t)(b * N_ + i)) * N_ + j];
            float mv  = (float)mask[((size_t)(b * N_ + i)) * N_ + j];
            float e   = (sRow[iLoc] + dCol[j] + coef * nrm * av) * mv;
            ebuf[iLoc][j] = e;
        }
    }
    __syncthreads();

    // Softmax: wave w owns rows 2w, 2w+1; 8 elements/lane, wave32 reductions.
#pragma unroll
    for (int rr = 0; rr < 2; ++rr) {
        const int iLoc = wave * 2 + rr;
        float vals[8];
        float vmax = -3.402823466e38f;
#pragma unroll
        for (int u = 0; u < 8; ++u) {
            vals[u] = ebuf[iLoc][lane + u * 32];
            vmax = fmaxf(vmax, vals[u]);
        }
#pragma unroll
        for (int o = 16; o > 0; o >>= 1) vmax = fmaxf(vmax, __shfl_xor(vmax, o, 32));
        float vsum = 0.f;
#pragma unroll
        for (int u = 0; u < 8; ++u) { vals[u] = __expf(vals[u] - vmax); vsum += vals[u]; }
#pragma unroll
        for (int o = 16; o > 0; o >>= 1) vsum += __shfl_xor(vsum, o, 32);
        const float inv = 1.f / vsum;
        float* orow = out + ((size_t)(b * N_ + iBase + iLoc)) * N_;
#pragma unroll
        for (int u = 0; u < 8; ++u) orow[lane + u * 32] = vals[u] * inv;
    }
}

// ---------------------------------------------------------------------------
extern "C" void kernel_launch(void* const* d_in, const int* in_sizes, int n_in,
                              void* d_out, int out_size, void* d_ws, size_t ws_size,
                              hipStream_t stream) {
    (void)in_sizes; (void)n_in; (void)out_size; (void)ws_size;
    const float* src    = (const float*)d_in[0];
    const int*   adj    = (const int*)  d_in[1];
    const int*   mask   = (const int*)  d_in[2];
    const float* W_lin  = (const float*)d_in[3];
    const float* a_src  = (const float*)d_in[4];
    const float* a_dst  = (const float*)d_in[5];
    const float* W_edge = (const float*)d_in[6];
    const float* a_edge = (const float*)d_in[7];
    float* out = (float*)d_out;

    float* ws   = (float*)d_ws;
    float* sArr = ws;
    float* dArr = ws + B_ * N_;
    float* qArr = ws + 2 * B_ * N_;

    gat_node_stats<<<dim3((B_ * N_) / 256), dim3(256), 0, stream>>>(
        src, W_lin, a_src, a_dst, sArr, dArr, qArr);

    gat_attn_softmax<<<dim3(N_ / 16, B_), dim3(256), 0, stream>>>(
        src, adj, mask, W_edge, a_edge, sArr, dArr, qArr, out);
}